// RGCN_61778809585719
// MI455X (gfx1250) — compile-verified
//
#include <hip/hip_runtime.h>
#include <hip/hip_bf16.h>

typedef float v2f __attribute__((ext_vector_type(2)));
typedef float v8f __attribute__((ext_vector_type(8)));
typedef unsigned int v4u __attribute__((ext_vector_type(4)));
typedef int v4i __attribute__((ext_vector_type(4)));
typedef int v8i __attribute__((ext_vector_type(8)));

#define N_NODES 50000
#define N_EDGES 400000
#define N_REL   3

// ---------------------------------------------------------------------------
// TDM: 2D tile DMA global -> LDS (CDNA5 Tensor Data Mover, ISA ch.8).
// D# group0: count=1, lds_addr, 57-bit global addr, type=2.
// D# group1: data_size=4B, tensor_dim0/1, tile_dim0/1, dim0 stride.
// Groups 2/3 zero (<=2D tensor). Issued once per wave; tracked by TENSORcnt.
// ---------------------------------------------------------------------------
__device__ __forceinline__ void tdm_load_2d(const void* gptr, unsigned lds_byte_off,
                                            unsigned tensor_d0, unsigned tensor_d1,
                                            unsigned tile_d0, unsigned tile_d1,
                                            unsigned long long stride0_elems) {
    unsigned long long ga = (unsigned long long)gptr;
    v4u g0;
    g0.x = 1u;                                            // count=1, is_restore=0
    g0.y = lds_byte_off;                                  // lds_addr [63:32]
    g0.z = (unsigned)(ga & 0xffffffffu);                  // global_addr lo
    g0.w = (unsigned)((ga >> 32) & 0x1ffffffu) | (2u << 30); // addr hi | type=2
    v8i g1;
    g1[0] = (int)(2u << 16);                              // data_size=2 (4 bytes)
    g1[1] = (int)((tensor_d0 & 0xffffu) << 16);           // tensor_dim0[15:0]
    g1[2] = (int)(((tensor_d0 >> 16) & 0xffffu) |
                  ((tensor_d1 & 0xffffu) << 16));         // dim0 hi | dim1 lo
    g1[3] = (int)(((tensor_d1 >> 16) & 0xffffu) |
                  ((tile_d0 & 0xffffu) << 16));           // dim1 hi | tile_dim0
    g1[4] = (int)(tile_d1 & 0xffffu);                     // tile_dim1 (tile_dim2=0)
    g1[5] = (int)(unsigned)(stride0_elems & 0xffffffffu); // tensor_dim0_stride lo
    g1[6] = (int)(unsigned)((stride0_elems >> 32) & 0xffffu); // stride hi (dim1 stride=0)
    g1[7] = 0;
    v4i z4 = {0, 0, 0, 0};
#if defined(__clang_major__) && (__clang_major__ >= 23)
    v8i z8 = {0, 0, 0, 0, 0, 0, 0, 0};
    __builtin_amdgcn_tensor_load_to_lds(g0, g1, z4, z4, z8, 0);
#else
    __builtin_amdgcn_tensor_load_to_lds(g0, g1, z4, z4, 0);
#endif
}

// ---------------------------------------------------------------------------
// Degree count: one thread per (relation, edge). Float atomics into deg arrays.
// ---------------------------------------------------------------------------
__global__ void degree_kernel(const int* __restrict__ edges,   // [3,2,E]
                              float* __restrict__ deg_out,     // [3,N]
                              float* __restrict__ deg_in,      // [3,N]
                              int E, int N) {
    int i = blockIdx.x * blockDim.x + threadIdx.x;
    if (i >= N_REL * E) return;
    int r = i / E;
    int e = i - r * E;
    int s = edges[(size_t)(r * 2 + 0) * E + e];
    int d = edges[(size_t)(r * 2 + 1) * E + e];
    atomicAdd(&deg_out[(size_t)r * N + s], 1.0f);
    atomicAdd(&deg_in [(size_t)r * N + d], 1.0f);
}

// In-place: deg -> rsqrt(max(deg, 1))
__global__ void rsqrt_kernel(float* __restrict__ p, int n) {
    int i = blockIdx.x * blockDim.x + threadIdx.x;
    if (i < n) p[i] = rsqrtf(fmaxf(p[i], 1.0f));
}

// ---------------------------------------------------------------------------
// WMMA f32 GEMM with TDM-staged LDS tiles:
//   H[M,Nout] = diag(rs) * X[M,K] @ W[K,Nout]
// Block = (Nout/16) waves, one 16x16 output tile per wave, blockIdx.x = M tile.
// Wave 0 DMAs the 16xK A tile (LDS off 0) and the full KxNout W panel
// (LDS off 16*K*4) via tensor_load_to_lds, waits TENSORcnt, then barrier.
// Inner loop reads fragments from LDS (ds_load) and chains
// V_WMMA_F32_16X16X4_F32 over K. M%16==0 (50000=3125*16) so EXEC is all-ones
// at every WMMA (ISA requirement; TDM/barrier branch reconverges first).
// ---------------------------------------------------------------------------
__global__ void wmma_gemm_scaled(const float* __restrict__ X,   // [M,K]
                                 const float* __restrict__ rs,  // [M]
                                 const float* __restrict__ W,   // [K,Nout]
                                 float* __restrict__ H,         // [M,Nout]
                                 int K, int Nout) {
    extern __shared__ float smem[];                 // [16*K] A | [K*Nout] W
    const int lane = threadIdx.x & 31;
    const int wave = threadIdx.x >> 5;
    const int m0   = blockIdx.x * 16;
    const int half = lane >> 4;   // 0: K pair {k,k+1}; 1: K pair {k+2,k+3}
    const int l    = lane & 15;

    if (threadIdx.x < 32) {
        // A tile: 16 rows x K cols, row stride K, into LDS offset 0
        tdm_load_2d(X + (size_t)m0 * K, 0u,
                    (unsigned)K, 16u, (unsigned)K, 16u, (unsigned long long)K);
        // W panel: K rows x Nout cols, row stride Nout, after the A tile
        tdm_load_2d(W, (unsigned)(16 * K * 4),
                    (unsigned)Nout, (unsigned)K, (unsigned)Nout, (unsigned)K,
                    (unsigned long long)Nout);
        __builtin_amdgcn_s_wait_tensorcnt(0);
    }
    __syncthreads();

    const float* __restrict__ As = smem;            // [16, K]
    const float* __restrict__ Ws = smem + 16 * K;   // [K, Nout]

    const float s   = rs[m0 + l];
    const int   n0  = wave * 16;
    const int   col = n0 + l;

    v8f acc = {};
    for (int k = 0; k < K; k += 4) {
        const int ka = k + half * 2;
        v2f a;
        a.x = As[l * K + ka]     * s;
        a.y = As[l * K + ka + 1] * s;
        v2f b;
        b.x = Ws[(ka)     * Nout + col];
        b.y = Ws[(ka + 1) * Nout + col];
        acc = __builtin_amdgcn_wmma_f32_16x16x4_f32(
            /*neg_a=*/false, a, /*neg_b=*/false, b,
            /*c_mod=*/(short)0, acc, /*reuse_a=*/false, /*reuse_b=*/false);
    }

#pragma unroll
    for (int v = 0; v < 8; ++v) {
        const int row = m0 + v + half * 8;          // C/D layout: lanes 16-31 -> M+8
        H[(size_t)row * Nout + col] = acc[v];
    }
}

// ---------------------------------------------------------------------------
// Edge scatter: one wave per edge. Each lane handles F/32 contiguous floats
// (vectorizes to b128/b64 loads). Message scaled by rsqrt(deg_in)[dst]
// before the atomic add so per-relation normalization is preserved.
// ---------------------------------------------------------------------------
template <int F>
__global__ void scatter_edges(const int* __restrict__ src,
                              const int* __restrict__ dst,
                              const float* __restrict__ H,      // [N,F]
                              const float* __restrict__ rs_in,  // [N]
                              float* __restrict__ acc,          // [N,F]
                              int E) {
    const int warp = blockIdx.x * (blockDim.x >> 5) + (threadIdx.x >> 5);
    if (warp >= E) return;
    const int lane = threadIdx.x & 31;
    const int s = src[warp];
    const int d = dst[warp];
    const float sc = rs_in[d];
    const float* __restrict__ hs = H   + (size_t)s * F;
    float* __restrict__       ad = acc + (size_t)d * F;
    constexpr int PER = F / 32;
#pragma unroll
    for (int i = 0; i < PER; ++i) {
        const int f = lane * PER + i;               // contiguous per lane
        atomicAdd(ad + f, hs[f] * sc);
    }
}

// acc = relu(acc + b[0] + b[1] + b[2]), F power of two.
__global__ void bias_relu_kernel(float* __restrict__ acc,
                                 const float* __restrict__ b,   // [3,F]
                                 size_t total, int F) {
    size_t i = (size_t)blockIdx.x * blockDim.x + threadIdx.x;
    if (i >= total) return;
    int f = (int)(i & (size_t)(F - 1));
    float bias = b[f] + b[F + f] + b[2 * F + f];
    acc[i] = fmaxf(acc[i] + bias, 0.0f);
}

// out = acc + b[0] + b[1] + b[2]
__global__ void bias_out_kernel(const float* __restrict__ acc,
                                const float* __restrict__ b,    // [3,F]
                                float* __restrict__ out,
                                size_t total, int F) {
    size_t i = (size_t)blockIdx.x * blockDim.x + threadIdx.x;
    if (i >= total) return;
    int f = (int)(i & (size_t)(F - 1));
    out[i] = acc[i] + b[f] + b[F + f] + b[2 * F + f];
}

// ---------------------------------------------------------------------------
extern "C" void kernel_launch(void* const* d_in, const int* in_sizes, int n_in,
                              void* d_out, int out_size, void* d_ws, size_t ws_size,
                              hipStream_t stream) {
    const float* x     = (const float*)d_in[0];   // [N,128]
    const int*   edges = (const int*)  d_in[1];   // [3,2,E]
    const float* W1    = (const float*)d_in[2];   // [3,128,128]
    const float* b1    = (const float*)d_in[3];   // [3,128]
    const float* W2    = (const float*)d_in[4];   // [3,128,64]
    const float* b2    = (const float*)d_in[5];   // [3,64]
    float*       out   = (float*)d_out;           // [N,64]

    const int N = N_NODES, E = N_EDGES;

    // Workspace layout (floats)
    float* ws     = (float*)d_ws;
    float* rs_out = ws;                              // [3,N]
    float* rs_in  = rs_out + (size_t)N_REL * N;      // [3,N]
    float* Hr     = rs_in  + (size_t)N_REL * N;      // [N,128] (reused both layers)
    float* acc1   = Hr     + (size_t)N * 128;        // [N,128]
    float* acc2   = acc1   + (size_t)N * 128;        // [N,64]

    // Zero accumulators / degree buffers (capture-safe async memsets)
    hipMemsetAsync(rs_out, 0, sizeof(float) * (size_t)2 * N_REL * N, stream);
    hipMemsetAsync(acc1,   0, sizeof(float) * (size_t)N * 128,       stream);
    hipMemsetAsync(acc2,   0, sizeof(float) * (size_t)N * 64,        stream);

    // Degrees -> rsqrt norms (shared by both layers)
    {
        int total = N_REL * E;
        degree_kernel<<<(total + 255) / 256, 256, 0, stream>>>(edges, rs_out, rs_in, E, N);
        int tn = 2 * N_REL * N;
        rsqrt_kernel<<<(tn + 255) / 256, 256, 0, stream>>>(rs_out, tn);
    }

    // ---- Layer 1: F_in=128 -> F_out=128, sum over relations, then bias+ReLU
    {
        const size_t shbytes = (size_t)(16 * 128 + 128 * 128) * sizeof(float); // 73728
        for (int r = 0; r < N_REL; ++r) {
            wmma_gemm_scaled<<<N / 16, (128 / 16) * 32, shbytes, stream>>>(
                x, rs_out + (size_t)r * N, W1 + (size_t)r * 128 * 128, Hr, 128, 128);
            const int* srcp = edges + (size_t)(r * 2 + 0) * E;
            const int* dstp = edges + (size_t)(r * 2 + 1) * E;
            scatter_edges<128><<<(E + 7) / 8, 256, 0, stream>>>(
                srcp, dstp, Hr, rs_in + (size_t)r * N, acc1, E);
        }
        size_t total = (size_t)N * 128;
        bias_relu_kernel<<<(unsigned)((total + 255) / 256), 256, 0, stream>>>(
            acc1, b1, total, 128);
    }

    // ---- Layer 2: F_in=128 -> F_out=64, sum over relations, then bias
    {
        const size_t shbytes = (size_t)(16 * 128 + 128 * 64) * sizeof(float); // 40960
        for (int r = 0; r < N_REL; ++r) {
            wmma_gemm_scaled<<<N / 16, (64 / 16) * 32, shbytes, stream>>>(
                acc1, rs_out + (size_t)r * N, W2 + (size_t)r * 128 * 64, Hr, 128, 64);
            const int* srcp = edges + (size_t)(r * 2 + 0) * E;
            const int* dstp = edges + (size_t)(r * 2 + 1) * E;
            scatter_edges<64><<<(E + 7) / 8, 256, 0, stream>>>(
                srcp, dstp, Hr, rs_in + (size_t)r * N, acc2, E);
        }
        size_t total = (size_t)N * 64;
        bias_out_kernel<<<(unsigned)((total + 255) / 256), 256, 0, stream>>>(
            acc2, b2, out, total, 64);
    }
}